// Decoder_HARNN_local_9964324127172
// MI455X (gfx1250) — compile-verified
//
#include <hip/hip_runtime.h>
#include <math.h>

#define DD   512
#define FCH  512
#define BB   32
#define SS   512
#define TOT  2798   // 9 + 128 + 661 + 2000

typedef __attribute__((ext_vector_type(2))) float v2f;
typedef __attribute__((ext_vector_type(8))) float v8f;

// ---------------------------------------------------------------------------
// local_input[b, 0:512] = lstm_pool[b, :]; local_input[b, 512:1024] = 0
// ---------------------------------------------------------------------------
__global__ void init_local_input(const float* __restrict__ lstm_pool,
                                 float* __restrict__ li) {
  int i = blockIdx.x * blockDim.x + threadIdx.x;   // 0 .. 32*1024-1
  if (i >= BB * 2 * DD) return;
  int b = i >> 10;
  int c = i & 1023;
  li[i] = (c < DD) ? lstm_pool[b * DD + c] : 0.0f;
}

// ---------------------------------------------------------------------------
// local_input[b, 512 + d] += sum over 64-seq chunk of lstm_out[b, s, d]
// grid (32, 8), block 128 threads; each thread owns a float4 column group.
// ---------------------------------------------------------------------------
__global__ void reduce_seq(const float* __restrict__ lstm_out,
                           float* __restrict__ li) {
  const int b     = blockIdx.x;     // 0..31
  const int chunk = blockIdx.y;     // 0..7
  const int t     = threadIdx.x;    // 0..127
  const float4* src =
      (const float4*)(lstm_out + (size_t)b * SS * DD + (size_t)chunk * 64 * DD);
  float4 acc = make_float4(0.f, 0.f, 0.f, 0.f);
#pragma unroll 4
  for (int s = 0; s < 64; ++s) {
    float4 v = src[s * (DD / 4) + t];
    acc.x += v.x; acc.y += v.y; acc.z += v.z; acc.w += v.w;
  }
  float* dst = li + b * (2 * DD) + DD + t * 4;
  atomicAdd(dst + 0, acc.x);
  atomicAdd(dst + 1, acc.y);
  atomicAdd(dst + 2, acc.z);
  atomicAdd(dst + 3, acc.w);
}

// ---------------------------------------------------------------------------
// fc = relu(local_input[32x1024] @ w_fc[1024x512] + b_fc)   (per level)
// One 16x16 tile per wave; 8 waves per block; grid = 8 blocks (64 tiles).
// ---------------------------------------------------------------------------
__global__ void fc_gemm(const float* __restrict__ A,     // [32 x 1024]
                        const float* __restrict__ Bw,    // [1024 x 512]
                        const float* __restrict__ bias,  // [512]
                        float* __restrict__ fc) {        // [32 x 512]
  const int lane  = threadIdx.x & 31;
  const int wave  = threadIdx.x >> 5;           // 0..7
  const int mt    = wave >> 2;                  // 0..1
  const int nt    = blockIdx.x * 4 + (wave & 3);// 0..31
  const int m     = lane & 15;
  const int khalf = (lane >> 4) * 2;            // 0 or 2
  const int col   = nt * 16 + (lane & 15);

  const float* Arow = A + (size_t)(mt * 16 + m) * (2 * DD);
  const float* Bcol = Bw + col;

  v8f acc = {};
#pragma unroll 4
  for (int k = 0; k < 2 * DD; k += 4) {
    v2f a = *(const v2f*)(Arow + k + khalf);            // global_load_b64
    v2f bb;
    bb.x = Bcol[(size_t)(k + khalf) * FCH];
    bb.y = Bcol[(size_t)(k + khalf + 1) * FCH];
    acc = __builtin_amdgcn_wmma_f32_16x16x4_f32(
        /*neg_a=*/false, a, /*neg_b=*/false, bb,
        /*c_mod=*/(short)0, acc, /*reuse_a=*/false, /*reuse_b=*/false);
  }

  const float bs = bias[col];
#pragma unroll
  for (int j = 0; j < 8; ++j) {
    int row = mt * 16 + j + (lane >> 4) * 8;
    float v = acc[j] + bs;
    fc[row * FCH + col] = v > 0.f ? v : 0.f;
  }
}

// ---------------------------------------------------------------------------
// scores = sigmoid(fc[32x512] @ w_lc[512xC] + b_lc)  -> out[:, off:off+C]
// N padded to 16; wave-uniform tile guard keeps EXEC all-1 at WMMA.
// ---------------------------------------------------------------------------
__global__ void score_gemm(const float* __restrict__ fc,   // [32 x 512]
                           const float* __restrict__ Bw,   // [512 x C]
                           const float* __restrict__ bias, // [C]
                           float* __restrict__ out,        // [32 x TOT]
                           int C, int off) {
  const int lane   = threadIdx.x & 31;
  const int wave   = threadIdx.x >> 5;
  const int ntiles = (C + 15) >> 4;
  const int t      = blockIdx.x * 8 + wave;     // wave-uniform
  if (t >= 2 * ntiles) return;                  // whole wave exits together
  const int mt    = t & 1;
  const int nt    = t >> 1;
  const int m     = lane & 15;
  const int khalf = (lane >> 4) * 2;
  const int col   = nt * 16 + (lane & 15);
  const bool inb  = (col < C);

  const float* Arow = fc + (size_t)(mt * 16 + m) * FCH;

  v8f acc = {};
#pragma unroll 4
  for (int k = 0; k < FCH; k += 4) {
    v2f a = *(const v2f*)(Arow + k + khalf);
    v2f bb;
    bb.x = inb ? Bw[(size_t)(k + khalf) * C + col] : 0.f;
    bb.y = inb ? Bw[(size_t)(k + khalf + 1) * C + col] : 0.f;
    acc = __builtin_amdgcn_wmma_f32_16x16x4_f32(
        false, a, false, bb, (short)0, acc, false, false);
  }

  if (inb) {
    const float bs = bias[col];
#pragma unroll
    for (int j = 0; j < 8; ++j) {
      int row = mt * 16 + j + (lane >> 4) * 8;
      float v = acc[j] + bs;
      out[row * TOT + off + col] = 1.0f / (1.0f + expf(-v));
    }
  }
}

// ---------------------------------------------------------------------------
// d_in order: 0 lstm_out, 1 lstm_pool, 2-5 w_at1, 6-9 w_at2,
//             10-13 w_fc, 14-17 b_fc, 18-21 w_lc, 22-25 b_lc
// (w_at1 / w_at2 provably do not affect the output: softmax sums to 1 over the
//  reduced axis, and mean-of-softmax is the constant 1/C.)
// ---------------------------------------------------------------------------
extern "C" void kernel_launch(void* const* d_in, const int* in_sizes, int n_in,
                              void* d_out, int out_size, void* d_ws, size_t ws_size,
                              hipStream_t stream) {
  (void)in_sizes; (void)n_in; (void)out_size; (void)ws_size;

  const float* lstm_out  = (const float*)d_in[0];
  const float* lstm_pool = (const float*)d_in[1];

  float* li = (float*)d_ws;                 // [32 x 1024]  (128 KB)
  float* fc = li + BB * 2 * DD;             // [32 x 512]   (64 KB, reused)
  float* out = (float*)d_out;               // [32 x 2798]

  // Build local_input = [lstm_pool | sum_s lstm_out]
  init_local_input<<<dim3((BB * 2 * DD + 255) / 256), dim3(256), 0, stream>>>(
      lstm_pool, li);
  reduce_seq<<<dim3(BB, 8), dim3(128), 0, stream>>>(lstm_out, li);

  static const int HIER[4] = {9, 128, 661, 2000};
  static const int OFFS[4] = {0, 9, 137, 798};
  for (int h = 0; h < 4; ++h) {
    const float* w_fc = (const float*)d_in[10 + h];
    const float* b_fc = (const float*)d_in[14 + h];
    const float* w_lc = (const float*)d_in[18 + h];
    const float* b_lc = (const float*)d_in[22 + h];

    fc_gemm<<<dim3(8), dim3(256), 0, stream>>>(li, w_fc, b_fc, fc);

    const int C = HIER[h];
    const int ntiles = (C + 15) >> 4;
    const int tiles  = 2 * ntiles;
    const int blocks = (tiles + 7) / 8;
    score_gemm<<<dim3(blocks), dim3(256), 0, stream>>>(fc, w_lc, b_lc, out, C,
                                                       OFFS[h]);
  }
}